// silutopsis_35124242546692
// MI455X (gfx1250) — compile-verified
//
#include <hip/hip_runtime.h>
#include <hip/hip_bf16.h>

// MI455X (gfx1250) implementation.
// Memory-bound problem: x traffic (205MB read x2 + 205MB write) dominates
// (~26us at 23.3 TB/s). Intermediates (12.8MB each) stay L2-resident.
// rect GEMM (3.3 GFLOP fp32) uses V_WMMA_F32_16X16X4_F32 for exact fp32.
// W is pre-packed into WMMA-fragment order so every B fragment is a single
// conflict-free ds_load_b64 (no VGPR repacking movs in the inner loop).

typedef __attribute__((ext_vector_type(2))) float v2f;
typedef __attribute__((ext_vector_type(8))) float v8f;

#define DIMC   512
#define NTOK   49
#define NROWS  6272          // 128 * 49 token rows
#define RCF    (NROWS * DIMC)

// ---------------- Kernel 1: 4x4 average pool, box[b*49+n][c] -----------------
__global__ __launch_bounds__(256) void pool16_k(const float* __restrict__ x,
                                                float* __restrict__ box) {
  int i  = blockIdx.x * 256 + threadIdx.x;       // total = 128*512*49 = 12544*256
  int n  = i % NTOK;
  int bc = i / NTOK;                             // b*512 + c
  int c  = bc & 511;
  int b  = bc >> 9;
  int py = n / 7, px = n % 7;
  const float* p = x + bc * 784 + py * 112 + px * 4;   // 16B aligned
  float s = 0.f;
#pragma unroll
  for (int r = 0; r < 4; ++r) {
    float4 v4 = *(const float4*)(p + r * 28);
    s += v4.x + v4.y + v4.z + v4.w;
  }
  box[(b * NTOK + n) * DIMC + c] = s * (1.f / 16.f);
}

// ------------- Kernel 2: 3x LayerNorm -> q,k,v (one wave per token) ----------
__global__ __launch_bounds__(256) void ln_qkv_k(const float* __restrict__ box,
    const float* __restrict__ qw, const float* __restrict__ qb,
    const float* __restrict__ kw, const float* __restrict__ kb,
    const float* __restrict__ vw, const float* __restrict__ vb,
    float* __restrict__ q, float* __restrict__ k, float* __restrict__ v) {
  int t    = threadIdx.x;
  int lane = t & 31;
  int w    = t >> 5;
  int row  = blockIdx.x * 8 + w;                 // < 6272
  const float* src = box + row * DIMC;
  float xv[16];
  float sum = 0.f, sq = 0.f;
#pragma unroll
  for (int i = 0; i < 16; ++i) {
    xv[i] = src[lane + (i << 5)];
    sum += xv[i];
    sq  += xv[i] * xv[i];
  }
  for (int off = 16; off > 0; off >>= 1) {
    sum += __shfl_xor(sum, off, 32);
    sq  += __shfl_xor(sq,  off, 32);
  }
  float m    = sum * (1.f / 512.f);
  float rstd = rsqrtf(sq * (1.f / 512.f) - m * m + 1e-6f);
#pragma unroll
  for (int i = 0; i < 16; ++i) {
    int c   = lane + (i << 5);
    float nv = (xv[i] - m) * rstd;
    q[row * DIMC + c] = nv * qw[c] + qb[c];
    k[row * DIMC + c] = nv * kw[c] + kb[c];
    v[row * DIMC + c] = nv * vw[c] + vb[c];
  }
}

// --- Kernel 3: pack W[j][k] into WMMA B-fragment order (one-time, 1MB) -------
// Wp[(k>>1)*1024 + j*2 + (k&1)] = W[j][k]  =>  per-lane B pair is contiguous.
__global__ __launch_bounds__(256) void packW_k(const float* __restrict__ W,
                                               float* __restrict__ Wp) {
  int i = blockIdx.x * 256 + threadIdx.x;        // 262144, coalesced read
  int j = i >> 9;
  int k = i & 511;
  Wp[(k >> 1) * 1024 + j * 2 + (k & 1)] = W[i];
}

// ------------- Kernel 4: attention per (batch, head), N=49, d=32 -------------
__global__ __launch_bounds__(256) void attn49_k(const float* __restrict__ q,
    const float* __restrict__ k, const float* __restrict__ v,
    float* __restrict__ shift) {
  __shared__ float qh[NTOK * 33];   // stride 33: avoid LDS bank conflicts
  __shared__ float kh[NTOK * 33];
  __shared__ float vh[NTOK * 33];
  __shared__ float sc[NTOK * 50];
  int t = threadIdx.x;
  int b = blockIdx.x >> 4;
  int g = blockIdx.x & 15;
  int base = (b * NTOK) * DIMC + g * 32;
  for (int idx = t; idx < NTOK * 32; idx += 256) {
    int n = idx >> 5, dd = idx & 31;
    int so = base + n * DIMC + dd;
    qh[n * 33 + dd] = q[so];
    kh[n * 33 + dd] = k[so];
    vh[n * 33 + dd] = v[so];
  }
  __syncthreads();
  const float scl = 0.17677669529663687f;        // 32^-0.5
  for (int e = t; e < NTOK * NTOK; e += 256) {
    int i = e / NTOK, j = e - i * NTOK;
    float acc = 0.f;
#pragma unroll 8
    for (int dd = 0; dd < 32; ++dd) acc += qh[i * 33 + dd] * kh[j * 33 + dd];
    sc[i * 50 + j] = acc * scl;
  }
  __syncthreads();
  if (t < NTOK) {
    float mx = -1e30f;
    for (int j = 0; j < NTOK; ++j) mx = fmaxf(mx, sc[t * 50 + j]);
    float sum = 0.f;
    for (int j = 0; j < NTOK; ++j) {
      float e2 = __expf(sc[t * 50 + j] - mx);
      sc[t * 50 + j] = e2;
      sum += e2;
    }
    float inv = 1.f / sum;
    for (int j = 0; j < NTOK; ++j) sc[t * 50 + j] *= inv;
  }
  __syncthreads();
  for (int e = t; e < NTOK * 32; e += 256) {
    int i = e >> 5, dd = e & 31;
    float acc = 0.f;
    for (int j = 0; j < NTOK; ++j) acc += sc[i * 50 + j] * vh[j * 33 + dd];
    shift[base + i * DIMC + dd] = acc;
  }
}

// -------- Kernel 5: Y = S @ W^T via V_WMMA_F32_16X16X4_F32, fused LN ---------
// Workgroup: 16 rows x 512 cols; 8 waves, each wave 16x64 = 4 C tiles.
// A staged in LDS (stride 514); packed-W staged in 16-K chunks, k-pair rows
// padded to 1056 dwords so lanes 0-15 use banks 0-31 and lanes 16-31 (k+2)
// use banks 32-63 -> conflict-free full-wave ds_load_b64 B fragments.
#define ASTR 514
#define PSTR 1056
#define YSTR 520
__global__ __launch_bounds__(256) void rect_gemm_ln_k(const float* __restrict__ S,
    const float* __restrict__ Wp, const float* __restrict__ lnw,
    const float* __restrict__ lnb, float* __restrict__ out) {
  __shared__ float sA[16 * ASTR];
  __shared__ float sB[8 * PSTR];                 // 8448 floats; also Y (16*520)
  __shared__ float mrow[16];
  __shared__ float rrow[16];
  int t       = threadIdx.x;
  int lane    = t & 31;
  int wave    = t >> 5;
  int rowbase = blockIdx.x * 16;
  int colbase = wave * 64;
  int mrw     = lane & 15;                        // A row / B col within tile
  int koff    = (lane >> 4) << 1;                 // lanes 16-31 hold K=2,3

  // stage A strip (16 x 512), coalesced float4, padded-stride LDS
  for (int idx = t * 4; idx < 8192; idx += 1024) {
    int r = idx >> 9, c = idx & 511;
    float4 v4 = *(const float4*)(S + rowbase * DIMC + idx);
    *(float2*)(&sA[r * ASTR + c])     = make_float2(v4.x, v4.y);
    *(float2*)(&sA[r * ASTR + c + 2]) = make_float2(v4.z, v4.w);
  }

  v8f acc[4] = {};
  for (int kb = 0; kb < DIMC; kb += 16) {
    // stage 8 packed k-pair rows (contiguous in Wp), float4 both sides
    for (int idx = t * 4; idx < 8192; idx += 1024) {
      int kp = idx >> 10, cc = idx & 1023;
      float4 v4 = *(const float4*)(Wp + (kb << 9) + idx);  // (kb/2)*1024
      *(float4*)(&sB[kp * PSTR + cc]) = v4;
    }
    __syncthreads();
#pragma unroll
    for (int kk = 0; kk < 16; kk += 4) {
      v2f a = *(const v2f*)(&sA[mrw * ASTR + kb + kk + koff]);
#pragma unroll
      for (int ct = 0; ct < 4; ++ct) {
        int n  = colbase + ct * 16 + mrw;
        int kp = (kk + koff) >> 1;
        v2f bfr = *(const v2f*)(&sB[kp * PSTR + n * 2]);   // single ds_load_b64
        acc[ct] = __builtin_amdgcn_wmma_f32_16x16x4_f32(
            false, a, false, bfr, (short)0, acc[ct], false, false);
      }
    }
    __syncthreads();
  }

  // scatter accumulators into sB as Y (16 x 512, stride YSTR)
  int mb = (lane >> 4) << 3;                      // lanes 16-31: M += 8
#pragma unroll
  for (int ct = 0; ct < 4; ++ct)
#pragma unroll
    for (int vv = 0; vv < 8; ++vv)
      sB[(vv + mb) * YSTR + colbase + ct * 16 + mrw] = acc[ct][vv];
  __syncthreads();

  // LayerNorm stats: 16 threads per row, 16-lane shuffle reduce
  {
    int r = t >> 4, j = t & 15;
    float sum = 0.f, sq = 0.f;
    for (int i = 0; i < 32; ++i) {
      float y = sB[r * YSTR + j + (i << 4)];
      sum += y;
      sq  += y * y;
    }
    for (int off = 8; off > 0; off >>= 1) {
      sum += __shfl_xor(sum, off, 16);
      sq  += __shfl_xor(sq,  off, 16);
    }
    if (j == 0) {
      float m   = sum * (1.f / 512.f);
      float var = sq * (1.f / 512.f) - m * m;
      mrow[r] = m;
      rrow[r] = rsqrtf(var + 1e-6f);
    }
  }
  __syncthreads();
  // normalized output, fully coalesced, token-major [row][c]
  for (int idx = t; idx < 8192; idx += 256) {
    int r = idx >> 9, c = idx & 511;
    float y = sB[r * YSTR + c];
    out[rowbase * DIMC + idx] = (y - mrow[r]) * rrow[r] * lnw[c] + lnb[c];
  }
}

// ------------- Kernel 6: broadcast SiLU gate, streaming float4 ---------------
__global__ __launch_bounds__(256) void gate_k(const float* __restrict__ x,
                                              const float* __restrict__ s,
                                              float* __restrict__ out) {
  int i  = blockIdx.x * 256 + threadIdx.x;       // one thread per 4x4 block
  int n  = i % NTOK;
  int bc = i / NTOK;
  int c  = bc & 511;
  int b  = bc >> 9;
  int py = n / 7, px = n % 7;
  float sh = s[(b * NTOK + n) * DIMC + c];
  int off  = bc * 784 + py * 112 + px * 4;
#pragma unroll
  for (int r = 0; r < 4; ++r) {
    float4 xv = *(const float4*)(x + off + r * 28);
    float4 o;
    o.x = xv.x / (1.f + __expf(-(sh + xv.x)));
    o.y = xv.y / (1.f + __expf(-(sh + xv.y)));
    o.z = xv.z / (1.f + __expf(-(sh + xv.z)));
    o.w = xv.w / (1.f + __expf(-(sh + xv.w)));
    *(float4*)(out + off + r * 28) = o;
  }
}

extern "C" void kernel_launch(void* const* d_in, const int* in_sizes, int n_in,
                              void* d_out, int out_size, void* d_ws, size_t ws_size,
                              hipStream_t stream) {
  const float* x   = (const float*)d_in[0];
  const float* qw  = (const float*)d_in[1];
  const float* qb  = (const float*)d_in[2];
  const float* kw  = (const float*)d_in[3];
  const float* kb  = (const float*)d_in[4];
  const float* vw  = (const float*)d_in[5];
  const float* vb  = (const float*)d_in[6];
  const float* rW  = (const float*)d_in[7];
  const float* lnw = (const float*)d_in[8];
  const float* lnb = (const float*)d_in[9];
  float* out = (float*)d_out;

  float* ws    = (float*)d_ws;
  float* box   = ws;                   // RCF
  float* q     = box + RCF;            // RCF
  float* k     = q + RCF;              // RCF
  float* v     = k + RCF;              // RCF
  float* shift = v + RCF;              // RCF
  float* Wp    = shift + RCF;          // 512*512 packed
  float* sOut  = Wp + DIMC * DIMC;     // RCF

  pool16_k      <<<12544, 256, 0, stream>>>(x, box);
  ln_qkv_k      <<<784,   256, 0, stream>>>(box, qw, qb, kw, kb, vw, vb, q, k, v);
  packW_k       <<<1024,  256, 0, stream>>>(rW, Wp);
  attn49_k      <<<2048,  256, 0, stream>>>(q, k, v, shift);
  rect_gemm_ln_k<<<392,   256, 0, stream>>>(shift, Wp, lnw, lnb, sOut);
  gate_k        <<<12544, 256, 0, stream>>>(x, sOut, out);
}